// AdaptiveReasoningAmplifier_28398323761450
// MI455X (gfx1250) — compile-verified
//
#include <hip/hip_runtime.h>

#define AMP_THRESHOLD 0.1f
#define CORR_THRESHOLD 0.3f
#define MAX_STEERING 0.5f
#define EPSF 1e-12f

typedef __attribute__((ext_vector_type(2))) float v2f;
typedef __attribute__((ext_vector_type(4))) float v4f;
typedef __attribute__((ext_vector_type(8))) float v8f;

#define WMMA_F32X4(A, B, C) \
    __builtin_amdgcn_wmma_f32_16x16x4_f32(false, (A), false, (B), (short)0, (C), false, false)

// ---------------------------------------------------------------------------
// Kernel 1: partial column sums of H[S x D] via V_WMMA_F32_16X16X4_F32.
// D-tile = ones[16x4] x B[4x16] + C => every row of D is the column sum of B.
// A is all-ones so the K-row->lane/VGPR mapping of B is irrelevant; only the
// N = lane&15 column mapping of C/D matters (documented layout).
// Manual 4x unroll: 8 independent b32 loads in flight before the wmma chain.
// ---------------------------------------------------------------------------
__global__ void colsum_wmma_kernel(const float* __restrict__ h,
                                   float* __restrict__ partial,
                                   int S, int D, int rpc) {
    const int lane     = threadIdx.x & 31;
    const int wave     = threadIdx.x >> 5;
    const int colGroup = blockIdx.x * (blockDim.x >> 5) + wave;
    if (colGroup * 16 >= D) return;          // wave-uniform (EXEC stays full)

    const int dcol  = colGroup * 16 + (lane & 15);
    const int g     = lane >> 4;             // half-wave picks 2 of 4 rows
    const int r0    = blockIdx.y * rpc;
    const int r1    = min(r0 + rpc, S);
    const int nrows = r1 - r0;

    const size_t sD   = (size_t)D;
    const size_t step = 4 * sD;              // 4 rows per wmma
    const float* p0 = h + (size_t)(r0 + g) * sD + dcol;   // rows r+{0,1}
    const float* p1 = p0 + 2 * sD;                        // rows r+{2,3}

    v2f a = {1.0f, 1.0f};
    v8f c = {0.f, 0.f, 0.f, 0.f, 0.f, 0.f, 0.f, 0.f};

    int quads = nrows >> 2;
    while (quads >= 4) {                     // 16 rows / iter, 8 loads in flight
        v2f b0, b1, b2, b3;
        b0.x = p0[0];        b0.y = p1[0];
        b1.x = p0[step];     b1.y = p1[step];
        b2.x = p0[2*step];   b2.y = p1[2*step];
        b3.x = p0[3*step];   b3.y = p1[3*step];
        p0 += 4 * step;  p1 += 4 * step;
        c = WMMA_F32X4(a, b0, c);
        c = WMMA_F32X4(a, b1, c);
        c = WMMA_F32X4(a, b2, c);
        c = WMMA_F32X4(a, b3, c);
        quads -= 4;
    }
    while (quads > 0) {
        v2f b;
        b.x = p0[0];  b.y = p1[0];
        p0 += step;   p1 += step;
        c = WMMA_F32X4(a, b, c);
        --quads;
    }

    float sum = c[0];                        // M=0 row of D = column sums
    const int  tail = nrows & 3;             // uniform; no divergence
    const float* pt = h + (size_t)(r1 - tail) * sD + dcol;
    for (int t = tail; t > 0; --t) { sum += pt[0]; pt += sD; }

    if (lane < 16)
        partial[(size_t)blockIdx.y * D + dcol] = sum;
}

// ---------------------------------------------------------------------------
// Kernel 2: reduce chunk partials + all scalar math (quality, alpha, norms).
// Writes [alpha, m_norm, inv_dn, scale].
// ---------------------------------------------------------------------------
__global__ void scalars_kernel(const float* __restrict__ partial,
                               const float* __restrict__ corr,
                               const float* __restrict__ incor,
                               const float* __restrict__ sscale,
                               float* __restrict__ scal,
                               int S, int D, int nChunks) {
    __shared__ float red[6][256];
    const int t = threadIdx.x;
    const float invS = 1.0f / (float)S;
    float cc = 0.f, cd = 0.f, ci = 0.f, nc = 0.f, ni = 0.f, dd = 0.f;
    for (int d = t; d < D; d += blockDim.x) {
        float cs = 0.f;
        for (int ch = 0; ch < nChunks; ++ch)       // fixed order => deterministic
            cs += partial[(size_t)ch * D + d];
        float cur = cs * invS;
        float co = corr[d], ic = incor[d];
        float df = co - ic;
        cc += cur * cur;  cd += cur * co;  ci += cur * ic;
        nc += co * co;    ni += ic * ic;   dd += df * df;
    }
    red[0][t] = cc; red[1][t] = cd; red[2][t] = ci;
    red[3][t] = nc; red[4][t] = ni; red[5][t] = dd;
    __syncthreads();
    for (int s = 128; s > 0; s >>= 1) {
        if (t < s)
            for (int k = 0; k < 6; ++k) red[k][t] += red[k][t + s];
        __syncthreads();
    }
    if (t == 0) {
        float cn   = sqrtf(red[0][0]);
        float csim = (sqrtf(red[3][0]) > 0.f) ? red[1][0] / fmaxf(cn, EPSF) : 0.f;
        float isim = (sqrtf(red[4][0]) > 0.f) ? red[2][0] / fmaxf(cn, EPSF) : 0.f;
        float quality = csim - isim;
        float alpha;
        if (quality < -CORR_THRESHOLD)      alpha = MAX_STEERING;
        else if (quality < AMP_THRESHOLD)   alpha = MAX_STEERING * (AMP_THRESHOLD - quality)
                                                    / (AMP_THRESHOLD + CORR_THRESHOLD);
        else                                alpha = 0.05f;
        float dn     = sqrtf(red[5][0]);
        float inv_dn = 1.0f / fmaxf(dn, EPSF);
        float m_norm = dn * inv_dn;          // ||momentum|| (1 unless degenerate)
        scal[0] = alpha;
        scal[1] = m_norm;
        scal[2] = inv_dn;
        scal[3] = sscale[0];
    }
}

// ---------------------------------------------------------------------------
// Kernel 3: out[s,d] = h[s,d] + coef(s) * (corr[d]-incor[d]) * inv_dn
// float4 vectorized. Output stores are non-temporal so the 128 MiB write
// stream doesn't evict h from L2 (h fits in the 192 MB L2 after pass 1).
// ---------------------------------------------------------------------------
__global__ void apply_kernel(const float* __restrict__ h,
                             const float* __restrict__ corr,
                             const float* __restrict__ incor,
                             const float* __restrict__ scal,
                             float* __restrict__ out,
                             int S, int D) {
    const int s = blockIdx.x;
    const int d = (blockIdx.y * blockDim.x + threadIdx.x) * 4;
    if (d + 3 >= D) return;

    const float alpha  = scal[0];
    const float m_norm = scal[1];
    const float inv_dn = scal[2];
    const float sc     = scal[3];

    float rel   = (float)s / (float)S;
    float eff   = alpha * (0.5f + 0.5f * rel) * sc;
    float dnorm = fabsf(eff) * m_norm;
    float coef  = (dnorm > MAX_STEERING) ? eff * (MAX_STEERING / dnorm) : eff;
    coef *= inv_dn;

    const size_t idx = (size_t)s * D + d;
    v4f hv = *(const v4f*)(h + idx);
    v4f cv = *(const v4f*)(corr + d);
    v4f iv = *(const v4f*)(incor + d);
    v4f ov;
    ov.x = fmaf(coef, cv.x - iv.x, hv.x);
    ov.y = fmaf(coef, cv.y - iv.y, hv.y);
    ov.z = fmaf(coef, cv.z - iv.z, hv.z);
    ov.w = fmaf(coef, cv.w - iv.w, hv.w);
    __builtin_nontemporal_store(ov, (v4f*)(out + idx));
}

extern "C" void kernel_launch(void* const* d_in, const int* in_sizes, int n_in,
                              void* d_out, int out_size, void* d_ws, size_t ws_size,
                              hipStream_t stream) {
    const float* h     = (const float*)d_in[0];
    const float* corr  = (const float*)d_in[1];
    const float* incor = (const float*)d_in[2];
    const float* scale = (const float*)d_in[3];
    float*       out   = (float*)d_out;

    const int D = in_sizes[1];
    const int S = in_sizes[0] / D;

    // Workspace layout: [chunks * D] partial column sums, then 4 scalars.
    size_t wsFloats = ws_size / sizeof(float);
    int nChunks = 1;
    if (wsFloats > (size_t)D + 8) {
        size_t maxCh = (wsFloats - 8) / (size_t)D;
        nChunks = (int)(maxCh < 64 ? maxCh : 64);
        if (nChunks < 1) nChunks = 1;
    }
    int rpc    = ((S + nChunks - 1) / nChunks + 3) & ~3;  // rows/chunk, mult of 4
    int chunks = (S + rpc - 1) / rpc;                     // actual chunks used

    float* partial = (float*)d_ws;
    float* scal    = partial + (size_t)chunks * D;

    const int wavesPerBlock = 8;                          // 256 threads = 8 wave32
    dim3 g1((D / 16 + wavesPerBlock - 1) / wavesPerBlock, chunks);
    colsum_wmma_kernel<<<g1, wavesPerBlock * 32, 0, stream>>>(h, partial, S, D, rpc);

    scalars_kernel<<<1, 256, 0, stream>>>(partial, corr, incor, scale, scal, S, D, chunks);

    dim3 g3(S, (D / 4 + 255) / 256);
    apply_kernel<<<g3, 256, 0, stream>>>(h, corr, incor, scal, out, S, D);
}